// LinkPredictionGNN_67791763800741
// MI455X (gfx1250) — compile-verified
//
#include <hip/hip_runtime.h>
#include <hip/hip_bf16.h>

typedef float v2f __attribute__((ext_vector_type(2)));
typedef float v8f __attribute__((ext_vector_type(8)));

#define N_NODES 100000
#define HID 64

// ---------------- degree / norm ----------------

__global__ void init_deg_kernel(float* __restrict__ deg, int n) {
    int i = blockIdx.x * blockDim.x + threadIdx.x;
    if (i < n) deg[i] = 1.0f;  // self loop
}

__global__ void degree_kernel(const int* __restrict__ dst, float* __restrict__ deg, int e) {
    int i = blockIdx.x * blockDim.x + threadIdx.x;
    if (i < e) atomicAdd(&deg[dst[i]], 1.0f);
}

__global__ void dinv_kernel(float* __restrict__ deg, int n) {
    int i = blockIdx.x * blockDim.x + threadIdx.x;
    if (i < n) {
        float d = deg[i];
        deg[i] = (d > 0.0f) ? rsqrtf(d) : 0.0f;  // in place: deg -> dinv
    }
}

// ---------------- WMMA GEMM: H[M,64] = A[M,K] @ W[K,64] ----------------
// One wave computes a 16x16 tile. Block = 128 threads = 4 waves covering N=64.
// grid.x = M/16 (M % 16 == 0 so EXEC is all-ones, required for WMMA).

template <int K>
__global__ __launch_bounds__(128) void gemm_wmma_kernel(const float* __restrict__ A,
                                                        const float* __restrict__ W,
                                                        float* __restrict__ H) {
    const int wave  = threadIdx.x >> 5;   // 0..3 -> N tile
    const int lane  = threadIdx.x & 31;
    const int l15   = lane & 15;
    const int khalf = (lane >> 4) << 1;   // 0 for lanes 0-15, 2 for lanes 16-31
    const int m0    = blockIdx.x << 4;
    const int n0    = wave << 4;

    v8f c = {0.f, 0.f, 0.f, 0.f, 0.f, 0.f, 0.f, 0.f};

    const float* arow = A + (size_t)(m0 + l15) * K;
    const float* wcol = W + n0 + l15;

#pragma unroll
    for (int k = 0; k < K; k += 4) {
        // A fragment: lane holds A[m0+l15][k+khalf], A[m0+l15][k+khalf+1]
        v2f a = *(const v2f*)(arow + k + khalf);
        // B fragment: lane holds W[k+khalf][n0+l15], W[k+khalf+1][n0+l15]
        v2f b;
        b.x = wcol[(k + khalf) * HID];
        b.y = wcol[(k + khalf + 1) * HID];
        c = __builtin_amdgcn_wmma_f32_16x16x4_f32(false, a, false, b, (short)0, c,
                                                  false, false);
    }

    // D layout: VGPR v -> row m0 + (lane>=16 ? 8 : 0) + v, col n0 + (lane&15)
    float* out = H + (size_t)(m0 + (lane >> 4) * 8) * HID + n0 + l15;
#pragma unroll
    for (int v = 0; v < 8; ++v) out[(size_t)v * HID] = c[v];
}

// ---------------- aggregation ----------------
// agg[i][c] = dinv[i]^2 * h[i][c] + bias[c]   (self-loop term + bias)

__global__ void self_init_kernel(const float* __restrict__ h,
                                 const float* __restrict__ dinv,
                                 const float* __restrict__ bias,
                                 float* __restrict__ agg, int n) {
    int t = blockIdx.x * blockDim.x + threadIdx.x;
    int i = t >> 4;
    if (i >= n) return;
    int c4 = (t & 15) << 2;
    float di = dinv[i];
    float w = di * di;
    float4 v = *(const float4*)(h + (size_t)i * HID + c4);
    float4 b = *(const float4*)(bias + c4);
    float4 o;
    o.x = v.x * w + b.x;
    o.y = v.y * w + b.y;
    o.z = v.z * w + b.z;
    o.w = v.w * w + b.w;
    *(float4*)(agg + (size_t)i * HID + c4) = o;
}

// Per edge: agg[dst] += dinv[src]*dinv[dst] * h[src]. 16 lanes per edge, 4 ch each.

__global__ void edge_scatter_kernel(const int* __restrict__ src,
                                    const int* __restrict__ dst,
                                    const float* __restrict__ h,
                                    const float* __restrict__ dinv,
                                    float* __restrict__ agg, int e) {
    int t = blockIdx.x * blockDim.x + threadIdx.x;
    int ed = t >> 4;
    if (ed >= e) return;
    int c4 = (t & 15) << 2;
    int s = src[ed];
    int d = dst[ed];
    float norm = dinv[s] * dinv[d];
    float4 v = *(const float4*)(h + (size_t)s * HID + c4);
    float* o = agg + (size_t)d * HID + c4;
    atomicAdd(o + 0, v.x * norm);
    atomicAdd(o + 1, v.y * norm);
    atomicAdd(o + 2, v.z * norm);
    atomicAdd(o + 3, v.w * norm);
}

__global__ void relu_kernel(float* __restrict__ a, int n4) {
    int t = blockIdx.x * blockDim.x + threadIdx.x;
    if (t >= n4) return;
    float4 v = *(const float4*)(a + (size_t)t * 4);
    v.x = fmaxf(v.x, 0.0f);
    v.y = fmaxf(v.y, 0.0f);
    v.z = fmaxf(v.z, 0.0f);
    v.w = fmaxf(v.w, 0.0f);
    *(float4*)(a + (size_t)t * 4) = v;
}

// ---------------- launch ----------------

extern "C" void kernel_launch(void* const* d_in, const int* in_sizes, int n_in,
                              void* d_out, int out_size, void* d_ws, size_t ws_size,
                              hipStream_t stream) {
    const float* x  = (const float*)d_in[0];
    const int*   ei = (const int*)d_in[1];
    const float* W1 = (const float*)d_in[2];
    const float* b1 = (const float*)d_in[3];
    const float* W2 = (const float*)d_in[4];
    const float* b2 = (const float*)d_in[5];
    float* out = (float*)d_out;

    const int M = N_NODES;
    const int E = in_sizes[1] / 2;
    const int* src = ei;
    const int* dst = ei + E;

    // workspace carve-up (512B aligned)
    char* ws = (char*)d_ws;
    float* dinv = (float*)(ws);                           // M floats (deg then dinv)
    float* hbuf = (float*)(ws + 400384);                  // M*64 floats
    float* agg  = (float*)(ws + 400384 + 25600000);       // M*64 floats

    const int B = 256;
    int gN  = (M + B - 1) / B;                // per-node
    int gE  = (E + B - 1) / B;                // per-edge
    int gN16 = (M * 16 + B - 1) / B;          // node x 16 lanes (float4 channels)
    int gE16 = (E * 16 + B - 1) / B;          // edge x 16 lanes
    int gM4  = (M * 16 + B - 1) / B;          // M*64/4 elements
    int gTile = M / 16;                       // 6250 WMMA row-tiles

    // degree / norm
    init_deg_kernel<<<gN, B, 0, stream>>>(dinv, M);
    degree_kernel<<<gE, B, 0, stream>>>(dst, dinv, E);
    dinv_kernel<<<gN, B, 0, stream>>>(dinv, M);

    // ---- layer 1 ----
    gemm_wmma_kernel<128><<<gTile, 128, 0, stream>>>(x, W1, hbuf);
    self_init_kernel<<<gN16, B, 0, stream>>>(hbuf, dinv, b1, agg, M);
    edge_scatter_kernel<<<gE16, B, 0, stream>>>(src, dst, hbuf, dinv, agg, E);
    relu_kernel<<<gM4, B, 0, stream>>>(agg, M * 16);

    // ---- layer 2 ----
    gemm_wmma_kernel<64><<<gTile, 128, 0, stream>>>(agg, W2, hbuf);
    self_init_kernel<<<gN16, B, 0, stream>>>(hbuf, dinv, b2, out, M);
    edge_scatter_kernel<<<gE16, B, 0, stream>>>(src, dst, hbuf, dinv, out, E);
}